// HeterogeneousMoE_58067957842025
// MI455X (gfx1250) — compile-verified
//
#include <hip/hip_runtime.h>
#include <math.h>

// ---- problem dims (compile-time) -------------------------------------------
#define BB   2
#define TT   512
#define DD   512
#define BT   1024          // B*T
#define EE   8
#define DIH  1024          // DI = 2*D
#define DSN  16            // mamba state size
#define DTRN 32
#define WINW 64

typedef __attribute__((ext_vector_type(2))) float v2f;
typedef __attribute__((ext_vector_type(4))) float v4f;
typedef __attribute__((ext_vector_type(8))) float v8f;

// ---- CDNA5 async-to-LDS feature probes --------------------------------------
#if defined(__has_builtin)
#if __has_builtin(__builtin_amdgcn_global_load_async_to_lds_b32)
#define HAVE_ASYNC_LDS 1
typedef __attribute__((address_space(1))) int as1_int;   // global
typedef __attribute__((address_space(3))) int as3_int;   // LDS
#endif
#if __has_builtin(__builtin_amdgcn_s_wait_asynccnt)
#define ASYNC_WAIT0() __builtin_amdgcn_s_wait_asynccnt(0)
#else
#define ASYNC_WAIT0() asm volatile("s_wait_asynccnt 0x0" ::: "memory")
#endif
#endif

__device__ __forceinline__ float gelu_f(float x) {
  float x3 = x * x * x;
  return 0.5f * x * (1.0f + tanhf(0.7978845608028654f * (x + 0.044715f * x3)));
}
__device__ __forceinline__ float silu_f(float x) { return x / (1.0f + __expf(-x)); }
__device__ __forceinline__ float softplus_f(float x) {
  return (x > 20.0f) ? x : log1pf(__expf(x));
}

// ---- tiny utility kernels ---------------------------------------------------
__global__ void k_zero(float* p, int n) {
  int i = blockIdx.x * blockDim.x + threadIdx.x;
  if (i < n) p[i] = 0.0f;
}
__global__ void k_copy4(const v4f* __restrict__ a, v4f* __restrict__ b, int n4) {
  int i = blockIdx.x * blockDim.x + threadIdx.x;
  if (i < n4) b[i] = a[i];
}

// ---- RMSNorm + per-row mean (for spectral entropy) --------------------------
__global__ __launch_bounds__(256) void k_rmsnorm(const float* __restrict__ x,
                                                 const float* __restrict__ nw,
                                                 float* __restrict__ xn,
                                                 float* __restrict__ v) {
  __shared__ float red[256];
  const int row = blockIdx.x;                       // 0..BT-1
  const float* xr = x + (size_t)row * DD;
  float a0 = xr[threadIdx.x];
  float a1 = xr[threadIdx.x + 256];
  red[threadIdx.x] = a0 * a0 + a1 * a1;
  __syncthreads();
  for (int s = 128; s > 0; s >>= 1) {
    if (threadIdx.x < s) red[threadIdx.x] += red[threadIdx.x + s];
    __syncthreads();
  }
  float rs = rsqrtf(red[0] / (float)DD + 1e-6f);
  __syncthreads();
  float y0 = a0 * rs * nw[threadIdx.x];
  float y1 = a1 * rs * nw[threadIdx.x + 256];
  xn[(size_t)row * DD + threadIdx.x] = y0;
  xn[(size_t)row * DD + threadIdx.x + 256] = y1;
  red[threadIdx.x] = y0 + y1;
  __syncthreads();
  for (int s = 128; s > 0; s >>= 1) {
    if (threadIdx.x < s) red[threadIdx.x] += red[threadIdx.x + s];
    __syncthreads();
  }
  if (threadIdx.x == 0) v[row] = red[0] / (float)DD;
}

// ---- serial cumsum over T (per batch; T=512 is trivial) ---------------------
__global__ void k_cumsum(const float* __restrict__ v, float* __restrict__ c1,
                         float* __restrict__ c2) {
  int b = blockIdx.x;
  float s1 = 0.f, s2 = 0.f;
  for (int t = 0; t < TT; t++) {
    float vv = v[b * TT + t];
    s1 += vv; s2 += vv * vv;
    c1[b * TT + t] = s1; c2[b * TT + t] = s2;
  }
}

__global__ void k_ent(const float* __restrict__ c1, const float* __restrict__ c2,
                      float* __restrict__ ent) {
  int i = blockIdx.x * blockDim.x + threadIdx.x;
  if (i >= BT) return;
  int b = i / TT, t = i % TT;
  float p1 = (t >= WINW) ? c1[b * TT + t - WINW] : 0.f;
  float p2 = (t >= WINW) ? c2[b * TT + t - WINW] : 0.f;
  float mu  = (c1[i] - p1) / (float)WINW;
  float mu2 = (c2[i] - p2) / (float)WINW;
  float var = mu2 - mu * mu;
  var = var > 0.f ? var : 0.f;
  ent[i] = (logf(var + 1e-6f) + 10.f) * 0.05f;
}

// ---- gating: logits (one wave32 per token) ---------------------------------
__global__ __launch_bounds__(256) void k_logits(const float* __restrict__ xn,
                                                const float* __restrict__ gw,
                                                const float* __restrict__ ew,
                                                const float* __restrict__ eb,
                                                const float* __restrict__ temp,
                                                const float* __restrict__ ent,
                                                float* __restrict__ logits) {
  int wave = (blockIdx.x * blockDim.x + threadIdx.x) >> 5;
  int lane = threadIdx.x & 31;
  if (wave >= BT) return;
  const float* xr = xn + (size_t)wave * DD;
  float inv_t = 1.0f / (fabsf(temp[0]) + 1e-6f);
  for (int e = 0; e < EE; e++) {
    float p = 0.f;
    const float* g = gw + e * DD;
    for (int d = lane; d < DD; d += 32) p += xr[d] * g[d];
    for (int off = 16; off; off >>= 1) p += __shfl_xor(p, off, 32);
    if (lane == 0) logits[wave * EE + e] = (p + ent[wave] * ew[e] + eb[e]) * inv_t;
  }
}

// ---- top-2 softmax weights + aux accumulators -------------------------------
__global__ void k_gate(const float* __restrict__ logits, float* __restrict__ wm,
                       float* __restrict__ auxacc) {
  int i = blockIdx.x * blockDim.x + threadIdx.x;
  if (i >= BT) return;
  float l[EE];
  for (int e = 0; e < EE; e++) l[e] = logits[i * EE + e];
  int e1 = 0;
  for (int e = 1; e < EE; e++) if (l[e] > l[e1]) e1 = e;
  int e2 = -1;
  for (int e = 0; e < EE; e++) {
    if (e == e1) continue;
    if (e2 < 0 || l[e] > l[e2]) e2 = e;
  }
  float d = __expf(l[e2] - l[e1]);
  float w1 = 1.0f / (1.0f + d);
  float w2 = d / (1.0f + d);
  for (int e = 0; e < EE; e++) wm[i * EE + e] = 0.f;
  wm[i * EE + e1] = w1;
  wm[i * EE + e2] = w2;
  float mx = l[e1], se = 0.f, sm[EE];
  for (int e = 0; e < EE; e++) { sm[e] = __expf(l[e] - mx); se += sm[e]; }
  for (int e = 0; e < EE; e++) atomicAdd(&auxacc[e], sm[e] / se);
  atomicAdd(&auxacc[EE + e1], 1.0f);
  atomicAdd(&auxacc[EE + e2], 1.0f);
}

__global__ void k_aux(const float* __restrict__ auxacc, float* __restrict__ out_aux) {
  float a = 0.f;
  for (int e = 0; e < EE; e++)
    a += (auxacc[EE + e] / (float)BT) * (auxacc[e] / (float)BT);
  out_aux[0] = (float)EE * a;
}

// ---- fp32 WMMA GEMM: Out[M,N] = epilogue(A[M,K] @ W[N,K]^T) -----------------
// One wave computes a 32(M) x 64(N) strip: 2x4 v8f accumulators (64 VGPRs).
// A fragments shared across the 4 N-subtiles, B fragments across 2 M-subtiles:
// 6 float2 loads feed 8 independent WMMAs per K=4 step.  Fragments for step k
// are loaded at step k-4 (software pipeline) so s_wait_loadcnt sinks a full
// WMMA group away from its loads.
// EPI: 0=raw(+bias) 1=gelu(+bias) 2=softplus(+bias) 3=(+bias)*rowscale, +=
#define WMMA4(A_, B_, C_)                                                      \
  __builtin_amdgcn_wmma_f32_16x16x4_f32(false, (A_), false, (B_), (short)0,    \
                                        (C_), false, false)

template <int EPI>
__global__ __launch_bounds__(256) void k_gemm(const float* __restrict__ A,
                                              const float* __restrict__ Bw,
                                              float* __restrict__ Out,
                                              const float* __restrict__ bias,
                                              const float* __restrict__ scale,
                                              int scale_stride,
                                              int M, int N, int K,
                                              int lda, int ldb, int ldo) {
  const int wave = (blockIdx.x * blockDim.x + threadIdx.x) >> 5;
  const int lane = threadIdx.x & 31;
  const int tiles_n = N >> 6;
  const int tiles_m = M >> 5;
  if (wave >= tiles_m * tiles_n) return;   // whole-wave uniform: EXEC all ones
  const int m0 = (wave / tiles_n) << 5;
  const int n0 = (wave % tiles_n) << 6;
  const int l16 = lane & 15;
  const int lh  = lane >> 4;

  v8f acc[2][4];
#pragma unroll
  for (int i = 0; i < 2; i++)
#pragma unroll
    for (int s = 0; s < 4; s++) acc[i][s] = (v8f){0, 0, 0, 0, 0, 0, 0, 0};

  const float* Ap0 = A + (size_t)(m0 + l16) * lda + 2 * lh;
  const float* Ap1 = Ap0 + (size_t)16 * lda;
  const float* Bp0 = Bw + (size_t)(n0 +  0 + l16) * ldb + 2 * lh;
  const float* Bp1 = Bw + (size_t)(n0 + 16 + l16) * ldb + 2 * lh;
  const float* Bp2 = Bw + (size_t)(n0 + 32 + l16) * ldb + 2 * lh;
  const float* Bp3 = Bw + (size_t)(n0 + 48 + l16) * ldb + 2 * lh;

  // pipeline prologue: fragments for k = 0
  v2f a0 = *(const v2f*)Ap0;
  v2f a1 = *(const v2f*)Ap1;
  v2f b0 = *(const v2f*)Bp0;
  v2f b1 = *(const v2f*)Bp1;
  v2f b2 = *(const v2f*)Bp2;
  v2f b3 = *(const v2f*)Bp3;

#pragma unroll 4
  for (int k = 4; k < K; k += 4) {
    // prefetch next step's fragments before consuming current ones
    v2f a0n = *(const v2f*)(Ap0 + k);
    v2f a1n = *(const v2f*)(Ap1 + k);
    v2f b0n = *(const v2f*)(Bp0 + k);
    v2f b1n = *(const v2f*)(Bp1 + k);
    v2f b2n = *(const v2f*)(Bp2 + k);
    v2f b3n = *(const v2f*)(Bp3 + k);
    acc[0][0] = WMMA4(a0, b0, acc[0][0]);
    acc[0][1] = WMMA4(a0, b1, acc[0][1]);
    acc[0][2] = WMMA4(a0, b2, acc[0][2]);
    acc[0][3] = WMMA4(a0, b3, acc[0][3]);
    acc[1][0] = WMMA4(a1, b0, acc[1][0]);
    acc[1][1] = WMMA4(a1, b1, acc[1][1]);
    acc[1][2] = WMMA4(a1, b2, acc[1][2]);
    acc[1][3] = WMMA4(a1, b3, acc[1][3]);
    a0 = a0n; a1 = a1n;
    b0 = b0n; b1 = b1n; b2 = b2n; b3 = b3n;
  }
  // pipeline epilogue: last K-step
  acc[0][0] = WMMA4(a0, b0, acc[0][0]);
  acc[0][1] = WMMA4(a0, b1, acc[0][1]);
  acc[0][2] = WMMA4(a0, b2, acc[0][2]);
  acc[0][3] = WMMA4(a0, b3, acc[0][3]);
  acc[1][0] = WMMA4(a1, b0, acc[1][0]);
  acc[1][1] = WMMA4(a1, b1, acc[1][1]);
  acc[1][2] = WMMA4(a1, b2, acc[1][2]);
  acc[1][3] = WMMA4(a1, b3, acc[1][3]);

#pragma unroll
  for (int i = 0; i < 2; i++) {
#pragma unroll
    for (int s = 0; s < 4; s++) {
#pragma unroll
      for (int j = 0; j < 8; j++) {
        int m = m0 + 16 * i + j + 8 * lh;    // C/D layout: vgpr j -> M=j (+8 hi)
        int n = n0 + 16 * s + l16;
        float vv = acc[i][s][j];
        if (bias) vv += bias[n];
        if (EPI == 1) vv = gelu_f(vv);
        if (EPI == 2) vv = softplus_f(vv);
        if (EPI == 3)
          Out[(size_t)m * ldo + n] += vv * scale[(size_t)m * scale_stride];
        else
          Out[(size_t)m * ldo + n] = vv;
      }
    }
  }
}

// ---- depthwise causal convs -------------------------------------------------
__global__ __launch_bounds__(256) void k_dwconv7_gelu(const float* __restrict__ h,
                                                      const float* __restrict__ w,
                                                      const float* __restrict__ cb,
                                                      float* __restrict__ o) {
  int i = blockIdx.x * blockDim.x + threadIdx.x;
  if (i >= BT * DIH) return;
  int c = i % DIH, row = i / DIH;
  int t = row % TT, base = row - t;
  float acc = cb[c];
#pragma unroll
  for (int j = 0; j < 7; j++) {
    int tt = t - 6 + j;
    if (tt >= 0) acc += h[(size_t)(base + tt) * DIH + c] * w[c * 7 + j];
  }
  o[(size_t)row * DIH + c] = gelu_f(acc);
}

__global__ __launch_bounds__(256) void k_dwconv4_silu(const float* __restrict__ uz,
                                                      const float* __restrict__ w,
                                                      const float* __restrict__ cb,
                                                      float* __restrict__ u2) {
  int i = blockIdx.x * blockDim.x + threadIdx.x;
  if (i >= BT * DIH) return;
  int c = i % DIH, row = i / DIH;
  int t = row % TT, base = row - t;
  float acc = cb[c];
#pragma unroll
  for (int j = 0; j < 4; j++) {
    int tt = t - 3 + j;
    if (tt >= 0) acc += uz[(size_t)(base + tt) * (2 * DIH) + c] * w[c * 4 + j];
  }
  u2[(size_t)row * DIH + c] = silu_f(acc);
}

// ---- selective scan: one lane per (b, channel) ------------------------------
// Per-timestep B/C broadcast (32 floats shared by the whole block) is
// double-buffered in LDS.  On toolchains exposing the CDNA5 async-to-LDS path,
// wave 0 issues GLOBAL_LOAD_ASYNC_TO_LDS_B32 for step t+1 while all waves
// compute step t, then completes it with s_wait_asynccnt 0 before the barrier.
__global__ __launch_bounds__(256) void k_scan(const float* __restrict__ delta,
                                              const float* __restrict__ u2,
                                              const float* __restrict__ proj,
                                              const float* __restrict__ uz,
                                              const float* __restrict__ Alog,
                                              const float* __restrict__ Dm,
                                              float* __restrict__ g) {
  __shared__ float BC[2][2 * DSN];
  int b = blockIdx.x >> 2;                      // 4 blocks per batch
  int d = ((blockIdx.x & 3) << 8) + threadIdx.x;
  float As[DSN];
#pragma unroll
  for (int s = 0; s < DSN; s++) As[s] = -__expf(Alog[d * DSN + s]);
  float Dd = Dm[d];
  float h[DSN];
#pragma unroll
  for (int s = 0; s < DSN; s++) h[s] = 0.f;

  const int PR = DTRN + 2 * DSN;                // proj row stride (64)
#ifdef HAVE_ASYNC_LDS
  if (threadIdx.x < 32) {                       // wave 0 only (uniform branch)
    __builtin_amdgcn_global_load_async_to_lds_b32(
        (as1_int*)(proj + (size_t)(b * TT) * PR + DTRN + threadIdx.x),
        (as3_int*)&BC[0][threadIdx.x], 0, 0);
    ASYNC_WAIT0();
  }
  __syncthreads();
#endif

  for (int t = 0; t < TT; t++) {
    int row = b * TT + t;
    int cur = t & 1;
#ifdef HAVE_ASYNC_LDS
    if (t + 1 < TT && threadIdx.x < 32) {
      __builtin_amdgcn_global_load_async_to_lds_b32(
          (as1_int*)(proj + (size_t)(row + 1) * PR + DTRN + threadIdx.x),
          (as3_int*)&BC[cur ^ 1][threadIdx.x], 0, 0);
    }
#else
    if (threadIdx.x < 2 * DSN)
      BC[cur][threadIdx.x] = proj[(size_t)row * PR + DTRN + threadIdx.x];
    __syncthreads();
#endif
    float dl = delta[(size_t)row * DIH + d];
    float uu = u2[(size_t)row * DIH + d];
    float y = 0.f;
#pragma unroll
    for (int s = 0; s < DSN; s++) {
      h[s] = __expf(dl * As[s]) * h[s] + dl * BC[cur][s] * uu;
      y += h[s] * BC[cur][DSN + s];
    }
    y += uu * Dd;
    float z = uz[(size_t)row * (2 * DIH) + DIH + d];
    g[(size_t)row * DIH + d] = y * silu_f(z);
#ifdef HAVE_ASYNC_LDS
    if (threadIdx.x < 32) ASYNC_WAIT0();        // t+1 prefetch complete
#endif
    __syncthreads();
  }
}

// ---- host-side launch -------------------------------------------------------
static inline int gemm_grid(int M, int N) {
  int waves = (M >> 5) * (N >> 6);              // 32x64 per wave
  return (waves + 7) / 8;                       // 8 waves (256 thr) per block
}

extern "C" void kernel_launch(void* const* d_in, const int* in_sizes, int n_in,
                              void* d_out, int out_size, void* d_ws, size_t ws_size,
                              hipStream_t stream) {
  const float* x        = (const float*)d_in[0];
  const float* norm_w   = (const float*)d_in[1];
  const float* gate_w   = (const float*)d_in[2];
  const float* ent_w    = (const float*)d_in[3];
  const float* ent_b    = (const float*)d_in[4];
  const float* temp     = (const float*)d_in[5];
  const float* cfi_w    = (const float*)d_in[6];
  const float* cfi_b    = (const float*)d_in[7];
  const float* cdw_w    = (const float*)d_in[8];
  const float* cdw_b    = (const float*)d_in[9];
  const float* cfo_w    = (const float*)d_in[10];
  const float* cfo_b    = (const float*)d_in[11];
  const float* min_w    = (const float*)d_in[12];
  const float* mcv_w    = (const float*)d_in[13];
  const float* mcv_b    = (const float*)d_in[14];
  const float* mxp_w    = (const float*)d_in[15];
  const float* mdt_w    = (const float*)d_in[16];
  const float* mdt_b    = (const float*)d_in[17];
  const float* mAlog    = (const float*)d_in[18];
  const float* mD       = (const float*)d_in[19];
  const float* mout_w   = (const float*)d_in[20];
  float* out = (float*)d_out;

  // workspace carve-up (floats)
  float* ws = (float*)d_ws;
  size_t off = 0;
  float* xn     = ws + off; off += (size_t)BT * DD;
  float* vmean  = ws + off; off += BT;
  float* c1     = ws + off; off += BT;
  float* c2     = ws + off; off += BT;
  float* ent    = ws + off; off += BT;
  float* logits = ws + off; off += (size_t)BT * EE;
  float* wm     = ws + off; off += (size_t)BT * EE;
  float* auxacc = ws + off; off += 2 * EE;
  float* proj   = ws + off; off += (size_t)BT * (DTRN + 2 * DSN);
  float* bufA   = ws + off; off += (size_t)BT * DIH;   // conv h / mamba delta
  float* bufB   = ws + off; off += (size_t)BT * DIH;   // conv h2 / mamba g
  float* uz     = ws + off; off += (size_t)BT * 2 * DIH;
  float* u2     = ws + off; off += (size_t)BT * DIH;
  (void)ws_size; (void)in_sizes; (void)n_in; (void)out_size;

  // --- gating path -----------------------------------------------------------
  k_zero<<<1, 32, 0, stream>>>(auxacc, 2 * EE);
  k_rmsnorm<<<BT, 256, 0, stream>>>(x, norm_w, xn, vmean);
  k_cumsum<<<BB, 1, 0, stream>>>(vmean, c1, c2);
  k_ent<<<(BT + 255) / 256, 256, 0, stream>>>(c1, c2, ent);
  k_logits<<<BT / 8, 256, 0, stream>>>(xn, gate_w, ent_w, ent_b, temp, ent, logits);
  k_gate<<<(BT + 255) / 256, 256, 0, stream>>>(logits, wm, auxacc);
  k_aux<<<1, 1, 0, stream>>>(auxacc, out + (size_t)BT * DD);
  k_copy4<<<(BT * DD / 4) / 256, 256, 0, stream>>>(        // residual
      (const v4f*)x, (v4f*)out, BT * DD / 4);

  // --- conv experts ----------------------------------------------------------
  for (int e = 0; e < 4; e++) {
    const float* fiw = cfi_w + (size_t)e * DIH * DD;
    const float* fib = cfi_b + (size_t)e * DIH;
    const float* cw  = cdw_w + (size_t)e * DIH * 7;
    const float* cb  = cdw_b + (size_t)e * DIH;
    const float* fow = cfo_w + (size_t)e * DD * DIH;
    const float* fob = cfo_b + (size_t)e * DD;
    // h = gelu(xn @ fiw^T + fib)
    k_gemm<1><<<gemm_grid(BT, DIH), 256, 0, stream>>>(
        xn, fiw, bufA, fib, nullptr, 0, BT, DIH, DD, DD, DD, DIH);
    // h2 = gelu(dwconv7(h) + cb)
    k_dwconv7_gelu<<<(BT * DIH) / 256, 256, 0, stream>>>(bufA, cw, cb, bufB);
    // out += (h2 @ fow^T + fob) * wm[:,e]
    k_gemm<3><<<gemm_grid(BT, DD), 256, 0, stream>>>(
        bufB, fow, out, fob, wm + e, EE, BT, DD, DIH, DIH, DIH, DD);
  }

  // --- mamba experts ---------------------------------------------------------
  for (int m = 0; m < 4; m++) {
    const float* inw = min_w  + (size_t)m * 2 * DIH * DD;
    const float* cw  = mcv_w  + (size_t)m * DIH * 4;
    const float* cb  = mcv_b  + (size_t)m * DIH;
    const float* xpw = mxp_w  + (size_t)m * (DTRN + 2 * DSN) * DIH;
    const float* dtw = mdt_w  + (size_t)m * DIH * DTRN;
    const float* dtb = mdt_b  + (size_t)m * DIH;
    const float* Al  = mAlog  + (size_t)m * DIH * DSN;
    const float* Dm_ = mD     + (size_t)m * DIH;
    const float* ow  = mout_w + (size_t)m * DD * DIH;
    // uz = xn @ inw^T   (u = cols [0,DIH), z = cols [DIH,2*DIH))
    k_gemm<0><<<gemm_grid(BT, 2 * DIH), 256, 0, stream>>>(
        xn, inw, uz, nullptr, nullptr, 0, BT, 2 * DIH, DD, DD, DD, 2 * DIH);
    // u2 = silu(dwconv4(u) + cb)
    k_dwconv4_silu<<<(BT * DIH) / 256, 256, 0, stream>>>(uz, cw, cb, u2);
    // proj = u2 @ xpw^T   (dt | B | C)
    k_gemm<0><<<gemm_grid(BT, DTRN + 2 * DSN), 256, 0, stream>>>(
        u2, xpw, proj, nullptr, nullptr, 0,
        BT, DTRN + 2 * DSN, DIH, DIH, DIH, DTRN + 2 * DSN);
    // delta = softplus(dt @ dtw^T + dtb)   (dt strided in proj, lda=64, K=32)
    k_gemm<2><<<gemm_grid(BT, DIH), 256, 0, stream>>>(
        proj, dtw, bufA, dtb, nullptr, 0,
        BT, DIH, DTRN, DTRN + 2 * DSN, DTRN, DIH);
    // g = scan(u2, delta, A, B, C, D) * silu(z)
    k_scan<<<BB * (DIH / 256), 256, 0, stream>>>(bufA, u2, proj, uz, Al, Dm_, bufB);
    // out += (g @ ow^T) * wm[:,4+m]
    k_gemm<3><<<gemm_grid(BT, DD), 256, 0, stream>>>(
        bufB, ow, out, nullptr, wm + 4 + m, EE, BT, DD, DIH, DIH, DIH, DD);
  }
}